// ExtractTensorPatches_74861279969530
// MI455X (gfx1250) — compile-verified
//
#include <hip/hip_runtime.h>
#include <stdint.h>

// ExtractTensorPatches: x (16,3,512,512) f32, window 16x16, stride 8 -> out (16,3969,3,16,16) f32.
// HBM-store-bound (~195 MB out); input (50 MB) stays resident in the 192 MB L2,
// so the 4x read amplification from overlapping windows is free.
//
// Pipeline per block (256 thr, 16 KB LDS):
//   gather phase : thread t owns patch-row (g = blk*16 + t/16, i = t%16);
//                  4x global_load_async_to_lds_b128 with offset:{0,16,32,48}
//                  (one address computation, INST_OFFSET advances both sides)
//                  -> LDS already in output order.
//   sync         : s_wait_asynccnt 0 (own gathers landed) + workgroup barrier.
//   store phase  : lane-contiguous async stores LDS->global, 512 B/wave/instr,
//                  th:TH_STORE_NT so the 195 MB write stream doesn't evict the
//                  L2-resident input.

#define NC   3
#define NHW  512
#define NPW  63
#define THREADS 256
#define G_PER_BLOCK 16            // (b,ph,pw,c) groups per block; 64 float4 each
#define NBLOCKS 11907             // 16*3969*3 / 16  (exact, no tail)

__global__ __launch_bounds__(THREADS)
void ExtractTensorPatches_async_kernel(const float* __restrict__ x,
                                       float* __restrict__ out) {
    __shared__ float4 stage[THREADS * 4];   // 16 KB, holds block's 1024 float4s

    const uint32_t t = threadIdx.x;

    // ---------------- gather phase: one 64B patch row per thread ----------------
    const uint32_t g  = blockIdx.x * (uint32_t)G_PER_BLOCK + (t >> 4); // ((b*63+ph)*63+pw)*3+c
    const uint32_t i  = t & 15u;                                       // row within window
    const uint32_t c  = g % 3u;
    const uint32_t r  = g / 3u;            // b*3969 + ph*63 + pw
    const uint32_t pw = r % (uint32_t)NPW;
    const uint32_t r2 = r / (uint32_t)NPW; // b*63 + ph
    const uint32_t ph = r2 % (uint32_t)NPW;
    const uint32_t b  = r2 / (uint32_t)NPW;

    const uint32_t row = ph * 8u + i;
    const uint32_t src = ((b * (uint32_t)NC + c) * (uint32_t)NHW + row) * (uint32_t)NHW
                         + pw * 8u;                                   // floats, 32B aligned

    const uint64_t gaddr = (uint64_t)(uintptr_t)(x + src);
    const uint32_t laddr = (uint32_t)(uintptr_t)stage + t * 64u;      // output-order slot

    // INST_OFFSET is added to BOTH the LDS and the global address -> j4*16 on each side.
    asm volatile("global_load_async_to_lds_b128 %0, %1, off"           :: "v"(laddr), "v"(gaddr) : "memory");
    asm volatile("global_load_async_to_lds_b128 %0, %1, off offset:16" :: "v"(laddr), "v"(gaddr) : "memory");
    asm volatile("global_load_async_to_lds_b128 %0, %1, off offset:32" :: "v"(laddr), "v"(gaddr) : "memory");
    asm volatile("global_load_async_to_lds_b128 %0, %1, off offset:48" :: "v"(laddr), "v"(gaddr) : "memory");

    // own gathers are in LDS ...
    asm volatile("s_wait_asynccnt 0" ::: "memory");
    // ... and everyone else's too (store lanes read other waves' slots)
    __syncthreads();

    // ---------------- store phase: fully lane-contiguous, NT stream -------------
    const uint64_t obase = (uint64_t)(uintptr_t)out
                         + ((uint64_t)blockIdx.x * 1024u + t) * 16u;
    const uint32_t lbase = (uint32_t)(uintptr_t)stage + t * 16u;

#pragma unroll
    for (int k = 0; k < 4; ++k) {
        const uint64_t ga = obase + (uint64_t)k * 4096u;   // 256 lanes * 16B per chunk
        const uint32_t la = lbase + (uint32_t)k * 4096u;
        asm volatile("global_store_async_from_lds_b128 %0, %1, off th:TH_STORE_NT"
                     :: "v"(ga), "v"(la) : "memory");
    }
    // s_endpgm waits all counters (ASYNCcnt drained before LDS/wave teardown)
}

extern "C" void kernel_launch(void* const* d_in, const int* in_sizes, int n_in,
                              void* d_out, int out_size, void* d_ws, size_t ws_size,
                              hipStream_t stream) {
    (void)in_sizes; (void)n_in; (void)d_ws; (void)ws_size; (void)out_size;
    const float* x = (const float*)d_in[0];
    float* out     = (float*)d_out;
    hipLaunchKernelGGL(ExtractTensorPatches_async_kernel,
                       dim3(NBLOCKS), dim3(THREADS), 0, stream, x, out);
}